// GCN_9929964388496
// MI455X (gfx1250) — compile-verified
//
#include <hip/hip_runtime.h>
#include <hip/hip_bf16.h>

typedef __attribute__((ext_vector_type(2))) float v2f;
typedef __attribute__((ext_vector_type(8))) float v8f;

#define N_NODES 50000
#define N_EDGES 800000
#define F_IN    128
#define HIDDEN  256
#define N_CLS   4

// ---------------- degree / norm ----------------
__global__ void k_init_deg(float* __restrict__ deg, int n) {
    int i = blockIdx.x * blockDim.x + threadIdx.x;
    if (i < n) deg[i] = 1.0f;   // self-loop
}

__global__ void k_count_deg(const int* __restrict__ dst, float* __restrict__ deg, int e) {
    int i = blockIdx.x * blockDim.x + threadIdx.x;
    if (i < e) atomicAdd(&deg[dst[i]], 1.0f);
}

__global__ void k_rsqrt(const float* __restrict__ deg, float* __restrict__ dis, int n) {
    int i = blockIdx.x * blockDim.x + threadIdx.x;
    if (i < n) dis[i] = rsqrtf(deg[i]);
}

// ---------------- layer-1 aggregation on raw features (128), float4 wide ----------------
__global__ void k_self_agg1(const float4* __restrict__ x4, const float* __restrict__ dis,
                            float4* __restrict__ agg4, int total /* N*32 */) {
    int i = blockIdx.x * blockDim.x + threadIdx.x;
    if (i >= total) return;
    int node = i >> 5;                         // 32 float4 chunks per node
    float d = dis[node];
    float w = d * d;
    float4 v = x4[i];
    v.x *= w; v.y *= w; v.z *= w; v.w *= w;
    agg4[i] = v;
}

__global__ void k_edge_agg1(const int* __restrict__ src, const int* __restrict__ dst,
                            const float4* __restrict__ x4, const float* __restrict__ dis,
                            float* __restrict__ agg, int total /* E*32 */) {
    int i = blockIdx.x * blockDim.x + threadIdx.x;
    if (i >= total) return;
    int e = i >> 5;                            // edge
    int q = i & 31;                            // 4-feature chunk
    int s = src[e], d = dst[e];
    float w = dis[s] * dis[d];
    float4 v = x4[s * (F_IN / 4) + q];         // coalesced b128 gather
    float* ap = agg + d * F_IN + q * 4;
    atomicAdd(ap + 0, w * v.x);
    atomicAdd(ap + 1, w * v.y);
    atomicAdd(ap + 2, w * v.z);
    atomicAdd(ap + 3, w * v.w);
}

// ------- GEMM1: h1 = relu(agg[50000x128] @ W1[128x256] + b1) -------
// One wave owns one 16-row stripe and FOUR 16-col tiles (A fragments loaded once per 4 WMMAs).
__global__ __launch_bounds__(256) void k_gemm1(const float* __restrict__ A,
                                               const float* __restrict__ W,
                                               const float* __restrict__ bias,
                                               float* __restrict__ H, int M) {
    const int NGRP = 4;                               // 4 n-tiles per wave, 4 groups cover 256
    int lane = threadIdx.x & 31;
    int wave = blockIdx.x * (blockDim.x >> 5) + (threadIdx.x >> 5);
    int mt = wave / NGRP, ng = wave % NGRP;
    if (mt * 16 >= M) return;                         // wave-uniform guard
    int half = lane >> 4;                             // 0: K=0,1   1: K=2,3
    int mn   = lane & 15;
    int m0 = mt * 16;
    int nbase = ng * 64;                              // this wave covers cols [nbase, nbase+64)

    v8f c0 = {}, c1 = {}, c2 = {}, c3 = {};
    const float* arow = A + (m0 + mn) * F_IN;
    for (int k = 0; k < F_IN; k += 4) {
        int kk = k + 2 * half;
        v2f a;
        a.x = arow[kk];
        a.y = arow[kk + 1];
        const float* w0 = W + kk * HIDDEN + nbase + mn;
        const float* w1 = W + (kk + 1) * HIDDEN + nbase + mn;
        v2f b;
        b.x = w0[0];  b.y = w1[0];
        c0 = __builtin_amdgcn_wmma_f32_16x16x4_f32(false, a, false, b, (short)0, c0, false, false);
        b.x = w0[16]; b.y = w1[16];
        c1 = __builtin_amdgcn_wmma_f32_16x16x4_f32(false, a, false, b, (short)0, c1, false, false);
        b.x = w0[32]; b.y = w1[32];
        c2 = __builtin_amdgcn_wmma_f32_16x16x4_f32(false, a, false, b, (short)0, c2, false, false);
        b.x = w0[48]; b.y = w1[48];
        c3 = __builtin_amdgcn_wmma_f32_16x16x4_f32(false, a, false, b, (short)0, c3, false, false);
    }
    v8f acc[4] = {c0, c1, c2, c3};
#pragma unroll
    for (int t = 0; t < 4; ++t) {
        int col = nbase + t * 16 + mn;
        float bv = bias[col];
#pragma unroll
        for (int r = 0; r < 8; ++r) {
            int row = m0 + r + 8 * half;
            float v = acc[t][r] + bv;
            H[row * HIDDEN + col] = v > 0.0f ? v : 0.0f;
        }
    }
}

// ------- GEMM2: t = h1[50000x256] @ W2[256x4]  (N padded 4 -> 16) -------
__global__ __launch_bounds__(256) void k_gemm2(const float* __restrict__ A,
                                               const float* __restrict__ W,
                                               float* __restrict__ T, int M) {
    int lane = threadIdx.x & 31;
    int wave = blockIdx.x * (blockDim.x >> 5) + (threadIdx.x >> 5);
    int mt = wave;
    if (mt * 16 >= M) return;                         // wave-uniform guard
    int half = lane >> 4;
    int mn   = lane & 15;
    int m0 = mt * 16;

    v8f c = {};
    const float* arow = A + (m0 + mn) * HIDDEN;
    for (int k = 0; k < HIDDEN; k += 4) {
        int kk = k + 2 * half;
        v2f a, b;
        a.x = arow[kk];
        a.y = arow[kk + 1];
        float bx = 0.0f, by = 0.0f;
        if (mn < N_CLS) {                             // reconverges before WMMA
            bx = W[kk * N_CLS + mn];
            by = W[(kk + 1) * N_CLS + mn];
        }
        b.x = bx; b.y = by;
        c = __builtin_amdgcn_wmma_f32_16x16x4_f32(false, a, false, b, (short)0, c, false, false);
    }
    if (mn < N_CLS) {
#pragma unroll
        for (int r = 0; r < 8; ++r) {
            int row = m0 + r + 8 * half;
            T[row * N_CLS + mn] = c[r];
        }
    }
}

// ---------------- layer-2 aggregation on 4 classes ----------------
__global__ void k_init_out(const float4* __restrict__ t4, const float* __restrict__ dis,
                           const float* __restrict__ b2, float4* __restrict__ out4, int n) {
    int i = blockIdx.x * blockDim.x + threadIdx.x;
    if (i >= n) return;                               // one thread per node
    float d = dis[i];
    float w = d * d;
    float4 v = t4[i];
    v.x = w * v.x + b2[0];
    v.y = w * v.y + b2[1];
    v.z = w * v.z + b2[2];
    v.w = w * v.w + b2[3];
    out4[i] = v;
}

__global__ void k_edge_agg2(const int* __restrict__ src, const int* __restrict__ dst,
                            const float4* __restrict__ t4, const float* __restrict__ dis,
                            float* __restrict__ out, int e) {
    int i = blockIdx.x * blockDim.x + threadIdx.x;
    if (i >= e) return;                               // one thread per edge
    int s = src[i], d = dst[i];
    float w = dis[s] * dis[d];
    float4 v = t4[s];                                 // b128 gather
    float* op = out + d * N_CLS;
    atomicAdd(op + 0, w * v.x);
    atomicAdd(op + 1, w * v.y);
    atomicAdd(op + 2, w * v.z);
    atomicAdd(op + 3, w * v.w);
}

extern "C" void kernel_launch(void* const* d_in, const int* in_sizes, int n_in,
                              void* d_out, int out_size, void* d_ws, size_t ws_size,
                              hipStream_t stream) {
    const float* x    = (const float*)d_in[0];                 // [50000,128]
    const int*   eidx = (const int*)d_in[1];                   // [2,800000] (int32 under jax x64-off)
    const float* W1   = (const float*)d_in[2];                 // [128,256]
    const float* b1   = (const float*)d_in[3];                 // [256]
    const float* W2   = (const float*)d_in[4];                 // [256,4]
    const float* b2   = (const float*)d_in[5];                 // [4]
    float* out = (float*)d_out;                                // [50000,4]

    const int* src = eidx;
    const int* dst = eidx + N_EDGES;

    // workspace layout (floats): deg | dis | agg(128/node) | h1(256/node) | t(4/node)
    float* ws  = (float*)d_ws;
    float* deg = ws;
    float* dis = deg + N_NODES;
    float* agg = dis + N_NODES;
    float* h1  = agg + (size_t)N_NODES * F_IN;
    float* t   = h1 + (size_t)N_NODES * HIDDEN;

    const int TB = 256;
    auto blocks = [](long long n, int tb) { return (int)((n + tb - 1) / tb); };

    // normalization coefficients
    k_init_deg <<<blocks(N_NODES, TB), TB, 0, stream>>>(deg, N_NODES);
    k_count_deg<<<blocks(N_EDGES, TB), TB, 0, stream>>>(dst, deg, N_EDGES);
    k_rsqrt    <<<blocks(N_NODES, TB), TB, 0, stream>>>(deg, dis, N_NODES);

    // layer-1 aggregation (on 128 input features, before the GEMM), float4-wide
    k_self_agg1<<<blocks((long long)N_NODES * (F_IN / 4), TB), TB, 0, stream>>>(
        (const float4*)x, dis, (float4*)agg, N_NODES * (F_IN / 4));
    k_edge_agg1<<<blocks((long long)N_EDGES * (F_IN / 4), TB), TB, 0, stream>>>(
        src, dst, (const float4*)x, dis, agg, N_EDGES * (F_IN / 4));

    // GEMM1 + bias + relu  (one wave per 16-row stripe x 64 cols; 12500 waves)
    {
        int waves = (N_NODES / 16) * 4;                        // 3125 * 4
        k_gemm1<<<blocks(waves, 8), TB, 0, stream>>>(agg, W1, b1, h1, N_NODES);
    }

    // GEMM2 (N=4 padded into one 16-wide tile per row block; 3125 waves)
    {
        int waves = N_NODES / 16;
        k_gemm2<<<blocks(waves, 8), TB, 0, stream>>>(h1, W2, t, N_NODES);
    }

    // layer-2 aggregation (on 4 classes, after the GEMM) + bias
    k_init_out <<<blocks(N_NODES, TB), TB, 0, stream>>>((const float4*)t, dis, b2,
                                                        (float4*)out, N_NODES);
    k_edge_agg2<<<blocks(N_EDGES, TB), TB, 0, stream>>>(src, dst, (const float4*)t, dis,
                                                        out, N_EDGES);
}